// PCESTool_83837761618202
// MI455X (gfx1250) — compile-verified
//
#include <hip/hip_runtime.h>
#include <math.h>

// ---------------------------------------------------------------------------
// PCES pipeline for MI455X (gfx1250, wave32).
//   Phase 1: per-pillar QKV projection via V_WMMA_F32_16X16X4_F32 (K=C=4),
//            collapsed per-point gating weight, in-wave stable top-k select.
//   Phase 2: strictly sequential pillar expansion (lax.scan semantics),
//            one wave per batch element, wave-local FPS via shfl_xor argmax.
//   Phase 3: copy S -> out, derive enhanced masks.
// ---------------------------------------------------------------------------

#define BB     4
#define HH     48
#define WWID   48
#define PP     32
#define CC     4
#define HID    64
#define PILL2D (HH * WWID)
#define NPILL  (BB * PILL2D)
#define NPTS   (NPILL * PP)          // 294,912 point slots
#define ROWPAD 65                    // 65 % 64 banks == 1 -> conflict-free rows
#define THRESH 0.7f
#define RSQRT8 0.35355339059327376f

typedef __attribute__((ext_vector_type(2))) float v2f;
typedef __attribute__((ext_vector_type(8))) float v8f;

// ===========================================================================
// Phase 1: one wave (32 threads) per pillar; lane == point index.
// ===========================================================================
__global__ __launch_bounds__(32) void pces_phase1(
    const float* __restrict__ pts,  // [B,H,W,P,C]
    const float* __restrict__ pos,  // [P,C]
    const float* __restrict__ Wq, const float* __restrict__ bq,
    const float* __restrict__ Wk, const float* __restrict__ bk,
    const float* __restrict__ Wv, const float* __restrict__ bv,
    const float* __restrict__ Ww, const float* __restrict__ bw,
    float* __restrict__ S,          // [B,H,W,P,C] working point cloud
    float* __restrict__ selp,       // [B,H,W,P,C] selected points (sorted)
    float* __restrict__ selw)       // [B,H,W,P]   selected weights (sorted)
{
    const int pillar = blockIdx.x;
    const int l      = threadIdx.x;           // 0..31

    __shared__ float xs[PP][CC];              // x = points + pos_encoding
    __shared__ float qkv[3][PP][ROWPAD];      // Q,K,V rows (bias included)
    __shared__ float wbuf[CC][HID];           // current 4x64 weight matrix
    __shared__ float bbuf[HID];               // current bias
    __shared__ float wWs[HID];                // Ww (64x1)

    // ---- load this lane's point; derive mask (masked points are exactly 0) ----
    const float* myp = pts + ((size_t)pillar * PP + l) * CC;
    const float op0 = myp[0], op1 = myp[1], op2 = myp[2], op3 = myp[3];
    const bool  msk = (op0 != 0.f) || (op1 != 0.f) || (op2 != 0.f) || (op3 != 0.f);

    xs[l][0] = op0 + pos[l * CC + 0];
    xs[l][1] = op1 + pos[l * CC + 1];
    xs[l][2] = op2 + pos[l * CC + 2];
    xs[l][3] = op3 + pos[l * CC + 3];
    wWs[l]      = Ww[l];
    wWs[l + 32] = Ww[l + 32];

    // ---- initialize S = pillar_points (re-done every launch; deterministic) ----
    float* Sp = S + ((size_t)pillar * PP + l) * CC;
    Sp[0] = op0; Sp[1] = op1; Sp[2] = op2; Sp[3] = op3;

    // ---- QKV projections via V_WMMA_F32_16X16X4_F32 --------------------------
    // A (16x4 f32) layout per ISA: lanes 0-15 M=row, VGPR0=K0 VGPR1=K1;
    //                              lanes 16-31 hold K2/K3.   B mirrored.
    const float* Wm[3] = { Wq, Wk, Wv };
    const float* Bm[3] = { bq, bk, bv };
    const int r  = l & 15;
    const int hi = l >> 4;

    for (int mtx = 0; mtx < 3; ++mtx) {
        __syncthreads();                      // previous wbuf readers done
        const float* Wsrc = Wm[mtx];
#pragma unroll
        for (int i = 0; i < 8; ++i) {         // 32 lanes x 8 = 256 = 4x64
            int idx = l * 8 + i;
            wbuf[idx >> 6][idx & 63] = Wsrc[idx];
        }
        bbuf[l]      = Bm[mtx][l];
        bbuf[l + 32] = Bm[mtx][l + 32];
        __syncthreads();

        v2f a;
#pragma unroll
        for (int t = 0; t < 2; ++t) {         // M tiles: points 0-15 / 16-31
            a.x = xs[t * 16 + r][2 * hi + 0];
            a.y = xs[t * 16 + r][2 * hi + 1];
#pragma unroll
            for (int j = 0; j < 4; ++j) {     // N tiles: hidden cols 16j..16j+15
                v2f bo;
                bo.x = wbuf[2 * hi + 0][j * 16 + r];
                bo.y = wbuf[2 * hi + 1][j * 16 + r];
                v8f c = {};
                // 8 args: (neg_a, A, neg_b, B, c_mod, C, reuse_a, reuse_b)
                c = __builtin_amdgcn_wmma_f32_16x16x4_f32(
                        false, a, false, bo, (short)0, c, false, false);
                const int col = j * 16 + r;
#pragma unroll
                for (int rr = 0; rr < 8; ++rr)   // D: lane half hi owns M = rr+8*hi
                    qkv[mtx][t * 16 + rr + 8 * hi][col] = c[rr] + bbuf[col];
            }
        }
    }
    __syncthreads();

    // ---- collapsed gating weight per point -----------------------------------
    // w_lin = (1/sqrt(8)) * sum_{n,m} (Ww_n . q_m) * (v_n . k_m)
    float acc = 0.f;
    for (int m = 0; m < 8; ++m) {
        float qm[8], km[8];
#pragma unroll
        for (int d = 0; d < 8; ++d) {
            qm[d] = qkv[0][l][m * 8 + d];
            km[d] = qkv[1][l][m * 8 + d];
        }
        for (int n = 0; n < 8; ++n) {
            float A = 0.f, Bv = 0.f;
#pragma unroll
            for (int e = 0; e < 8; ++e) {
                A  += wWs[n * 8 + e]        * qm[e];   // Ww broadcast (uniform addr)
                Bv += qkv[2][l][n * 8 + e]  * km[e];   // v row, conflict-free (pad 65)
            }
            acc += A * Bv;
        }
    }
    const float z   = acc * RSQRT8 + bw[0];
    const float sig = 1.f / (1.f + expf(-z));
    const float wgt = msk ? sig : 0.f;                 // weights = sigmoid * mask

    // ---- stable descending top-k (Ksel == P == 32) ---------------------------
    const bool  sel   = msk && (wgt > THRESH);
    const float score = sel ? wgt : -1.f;
    int rank = 0;
    for (int j = 0; j < 32; ++j) {
        float sj = __shfl(score, j);
        rank += (sj > score) || ((sj == score) && (j < l));   // JAX tie-break
    }
    const unsigned long long selball  = __ballot(sel);
    const unsigned long long maskball = __ballot(msk);
    const int  count    = (int)__popcll(selball);
    const bool hasvalid = (maskball != 0ull);
    const int  first    = hasvalid ? (__ffsll((long long)maskball) - 1) : 0;
    const bool fallback = (count == 0) && hasvalid;

    const size_t sb   = (size_t)pillar * PP;
    const int    slot = fallback ? l : rank;           // fallback: rank==lane anyway
    float s0 = 0.f, s1 = 0.f, s2 = 0.f, s3 = 0.f, sw0 = 0.f;
    if (!fallback && sel) { s0 = op0; s1 = op1; s2 = op2; s3 = op3; sw0 = wgt; }
    selp[(sb + slot) * CC + 0] = s0;
    selp[(sb + slot) * CC + 1] = s1;
    selp[(sb + slot) * CC + 2] = s2;
    selp[(sb + slot) * CC + 3] = s3;
    selw[sb + slot] = sw0;
    if (fallback && (l == first)) {                    // overwrite slot 0 (ordered)
        selp[sb * CC + 0] = op0;
        selp[sb * CC + 1] = op1;
        selp[sb * CC + 2] = op2;
        selp[sb * CC + 3] = op3;
        selw[sb] = wgt;
    }
}

// ===========================================================================
// Phase 2: sequential expansion. One wave per batch element (scan carries S).
// ===========================================================================
__global__ __launch_bounds__(32) void pces_expand(
    float* __restrict__ S,
    const float* __restrict__ selp,
    const float* __restrict__ selw)
{
    const int b = blockIdx.x;
    const int l = threadIdx.x;

    float*       Sb = S    + (size_t)b * PILL2D * PP * CC;
    const float* sp = selp + (size_t)b * PILL2D * PP * CC;
    const float* sw = selw + (size_t)b * PILL2D * PP;

    __shared__ float cp[2 * PP][CC];   // compacted candidates
    __shared__ float cm[2 * PP];       // compacted validity
    __shared__ float np_[PP][CC];      // FPS output points

    const int dho[8] = { -1, -1, -1,  0, 0,  1, 1, 1 };
    const int dwo[8] = { -1,  0,  1, -1, 1, -1, 0, 1 };

    const unsigned long long ltm = (1ull << l) - 1ull;

    for (int p = 0; p < PILL2D; ++p) {
        const int h = p / WWID, w = p % WWID;

        const float swt = sw[(size_t)p * PP + l];
        const bool  cur = swt > 0.f;
        float vx = 0.f, vy = 0.f, vz = 0.f, vw = 0.f;
        if (cur) {
            const float* q = sp + ((size_t)p * PP + l) * CC;
            vx = q[0]; vy = q[1]; vz = q[2]; vw = q[3];
        }
        const unsigned long long cb = __ballot(cur);
        if (cb == 0ull) continue;                      // any_valid == false

        for (int o = 0; o < 8; ++o) {
            const int nh = h + dho[o], nw = w + dwo[o];
            if (nh < 0 || nh >= HH || nw < 0 || nw >= WWID) continue;
            float* Sn = Sb + ((size_t)(nh * WWID + nw)) * PP * CC;

            const float n0 = Sn[l * CC + 0], n1 = Sn[l * CC + 1];
            const float n2 = Sn[l * CC + 2], n3 = Sn[l * CC + 3];
            const bool  nm = ((n0 + n1 + n2 + n3) != 0.f);  // ref: sum != 0
            const unsigned long long nb = __ballot(nm);
            const int nvN     = (int)__popcll(nb);
            const int nvC     = (int)__popcll(cb);
            const int n_valid = nvN + nvC;
            const int nInvN   = PP - nvN;

            // stable compaction: valid first (neigh then vpts), invalid after
            const int dA = nm  ? (int)__popcll(nb & ltm)
                               : n_valid + (int)__popcll((~nb) & ltm);
            const int dB = cur ? nvN + (int)__popcll(cb & ltm)
                               : n_valid + nInvN + (int)__popcll((~cb) & ltm);
            cp[dA][0] = n0; cp[dA][1] = n1; cp[dA][2] = n2; cp[dA][3] = n3;
            cm[dA]    = nm ? 1.f : 0.f;
            cp[dB][0] = vx; cp[dB][1] = vy; cp[dB][2] = vz; cp[dB][3] = vw;
            cm[dB]    = cur ? 1.f : 0.f;
            __syncthreads();

            if (n_valid > PP) {
                // ---- farthest point sampling over 64 candidates ------------
                const float pAx = cp[l][0],      pAy = cp[l][1],      pAz = cp[l][2];
                const float pBx = cp[l + 32][0], pBy = cp[l + 32][1], pBz = cp[l + 32][2];
                const bool  mA  = cm[l] > 0.f,   mB  = cm[l + 32] > 0.f;
                float dAq = mA ? 1e10f : -__builtin_inff();
                float dBq = mB ? 1e10f : -__builtin_inff();
                int far = 0;
                for (int s = 0; s < PP; ++s) {
                    const float cx = cp[far][0], cy = cp[far][1];
                    const float cz = cp[far][2], cw = cp[far][3];
                    if (l == 0) { np_[s][0] = cx; np_[s][1] = cy;
                                  np_[s][2] = cz; np_[s][3] = cw; }
                    if (mA) {
                        const float t0 = pAx - cx, t1 = pAy - cy, t2 = pAz - cz;
                        dAq = fminf(dAq, t0 * t0 + t1 * t1 + t2 * t2);
                    }
                    if (mB) {
                        const float t0 = pBx - cx, t1 = pBy - cy, t2 = pBz - cz;
                        dBq = fminf(dBq, t0 * t0 + t1 * t1 + t2 * t2);
                    }
                    // argmax, lowest index wins ties (jnp.argmax semantics)
                    float bd = dAq; int bi = l;
                    if (dBq > bd) { bd = dBq; bi = l + 32; }
#pragma unroll
                    for (int off = 16; off > 0; off >>= 1) {
                        const float od = __shfl_xor(bd, off);
                        const int   oi = __shfl_xor(bi, off);
                        if (od > bd || (od == bd && oi < bi)) { bd = od; bi = oi; }
                    }
                    far = bi;                                  // wave-uniform
                }
                __syncthreads();
                Sn[l * CC + 0] = np_[l][0];
                Sn[l * CC + 1] = np_[l][1];
                Sn[l * CC + 2] = np_[l][2];
                Sn[l * CC + 3] = np_[l][3];
            } else {
                // direct: first P compacted entries, invalid zeroed
                const float m = cm[l];
                Sn[l * CC + 0] = (m > 0.f) ? cp[l][0] : 0.f;
                Sn[l * CC + 1] = (m > 0.f) ? cp[l][1] : 0.f;
                Sn[l * CC + 2] = (m > 0.f) ? cp[l][2] : 0.f;
                Sn[l * CC + 3] = (m > 0.f) ? cp[l][3] : 0.f;
            }
            __syncthreads();   // cp/cm reuse next offset; S store->load in-order
        }
    }
}

// ===========================================================================
// Phase 3: expanded -> d_out, enhanced_masks = (sum over C != 0)
// ===========================================================================
__global__ void pces_finalize(const float* __restrict__ S, float* __restrict__ out)
{
    const int i = blockIdx.x * blockDim.x + threadIdx.x;   // point slot
    if (i >= NPTS) return;
    const float a = S[i * CC + 0], b = S[i * CC + 1];
    const float c = S[i * CC + 2], d = S[i * CC + 3];
    out[i * CC + 0] = a; out[i * CC + 1] = b;
    out[i * CC + 2] = c; out[i * CC + 3] = d;
    out[(size_t)NPTS * CC + i] = ((a + b + c + d) != 0.f) ? 1.f : 0.f;
}

// ===========================================================================
extern "C" void kernel_launch(void* const* d_in, const int* in_sizes, int n_in,
                              void* d_out, int out_size, void* d_ws, size_t ws_size,
                              hipStream_t stream)
{
    // setup_inputs order:
    // 0 pillar_points f32, 1 pillar_masks (bool; unused -- masked points are
    //   exactly zero, so mask is derived as any(coord != 0)),
    // 2 pos_encoding, 3 Wq, 4 bq, 5 Wk, 6 bk, 7 Wv, 8 bv, 9 Ww, 10 bw
    const float* pts = (const float*)d_in[0];
    const float* pos = (const float*)d_in[2];
    const float* Wq  = (const float*)d_in[3];
    const float* bq  = (const float*)d_in[4];
    const float* Wk  = (const float*)d_in[5];
    const float* bk  = (const float*)d_in[6];
    const float* Wv  = (const float*)d_in[7];
    const float* bv  = (const float*)d_in[8];
    const float* Ww  = (const float*)d_in[9];
    const float* bw  = (const float*)d_in[10];
    float* out = (float*)d_out;

    // workspace: S | selp | selw  (~10.6 MB total)
    float* S    = (float*)d_ws;
    float* selp = S    + (size_t)NPTS * CC;
    float* selw = selp + (size_t)NPTS * CC;

    pces_phase1<<<NPILL, 32, 0, stream>>>(pts, pos, Wq, bq, Wk, bk, Wv, bv,
                                          Ww, bw, S, selp, selw);
    pces_expand<<<BB, 32, 0, stream>>>(S, selp, selw);
    pces_finalize<<<(NPTS + 255) / 256, 256, 0, stream>>>(S, out);

    (void)in_sizes; (void)n_in; (void)out_size; (void)ws_size;
}